// GAT_40467181863042
// MI455X (gfx1250) — compile-verified
//
#include <hip/hip_runtime.h>
#include <math.h>

typedef __attribute__((ext_vector_type(2))) float v2f;
typedef __attribute__((ext_vector_type(8))) float v8f;

#define FDIM 256
#define NEG_SLOPE 0.2f
#define EPSV 1e-16f

// ---------------------------------------------------------------- utilities

__global__ void fill_f32(float* __restrict__ p, float v, long long n) {
  long long i = (long long)blockIdx.x * blockDim.x + threadIdx.x;
  if (i < n) p[i] = v;
}

// Wt[n*K + k] = W[k*Nc + n]  (W: KxNc row-major -> Wt: NcxK row-major)
__global__ void transpose_f32(const float* __restrict__ W, float* __restrict__ Wt,
                              int K, int Nc) {
  int i = blockIdx.x * blockDim.x + threadIdx.x;
  if (i >= K * Nc) return;
  int k = i / Nc, n = i - k * Nc;
  Wt[(size_t)n * K + k] = W[i];
}

__device__ __forceinline__ void atomicMaxFloat(float* addr, float val) {
  // signed-int max for val>=0, unsigned min for val<0; lattice converges to fmax
  if (val >= 0.0f)
    atomicMax((int*)addr, __float_as_int(val));
  else
    atomicMin((unsigned int*)addr, __float_as_uint(val));
}

// ------------------------------------------------- fp32 WMMA GEMM  C = A @ B
// A: MxK row-major, Bt: NcxK row-major (pre-transposed B), C: MxNc row-major.
// M % 16 == 0, Nc % (16*NT) == 0, K % 4 == 0, K >= 8.
// One wave computes a 16 x (16*NT) output strip with V_WMMA_F32_16X16X4_F32.
// Software-pipelined: loads for k+4 are issued before the WMMAs consuming k,
// so WMMAs wait only on loads from the previous iteration.
template <int NT>
__global__ void gemm_f32_wmma(const float* __restrict__ A,
                              const float* __restrict__ Bt,
                              float* __restrict__ C,
                              int M, int K, int Nc) {
  int lane = threadIdx.x & 31;
  int wave = threadIdx.x >> 5;
  int grp  = blockIdx.x * (blockDim.x >> 5) + wave;
  int ngn  = Nc / (16 * NT);
  int total = (M >> 4) * ngn;
  if (grp >= total) return;               // wave-uniform: EXEC stays all-1s
  int mt = grp / ngn;
  int ng = grp - mt * ngn;

  int mrow = (mt << 4) + (lane & 15);     // A row owned by this lane
  int kb   = (lane >> 4) << 1;            // lanes 0-15 -> K0,K1; 16-31 -> K2,K3

  const float* Arow = A + (size_t)mrow * K + kb;
  const float* Brow[NT];
#pragma unroll
  for (int t = 0; t < NT; ++t) {
    int ncol = ((ng * NT + t) << 4) + (lane & 15);
    Brow[t] = Bt + (size_t)ncol * K + kb;
  }

  v8f z = {};
  v8f acc[NT];
#pragma unroll
  for (int t = 0; t < NT; ++t) acc[t] = z;

  // prologue: stage k = 0
  v2f a_cur = *(const v2f*)(Arow);
  v2f b_cur[NT];
#pragma unroll
  for (int t = 0; t < NT; ++t) b_cur[t] = *(const v2f*)(Brow[t]);

  for (int k = 0; k < K - 4; k += 4) {
    // issue next iteration's loads first (overlap with WMMAs below)
    v2f a_nxt = *(const v2f*)(Arow + k + 4);
    v2f b_nxt[NT];
#pragma unroll
    for (int t = 0; t < NT; ++t) b_nxt[t] = *(const v2f*)(Brow[t] + k + 4);

#pragma unroll
    for (int t = 0; t < NT; ++t)
      acc[t] = __builtin_amdgcn_wmma_f32_16x16x4_f32(
          false, a_cur, false, b_cur[t], (short)0, acc[t], false, false);

    a_cur = a_nxt;
#pragma unroll
    for (int t = 0; t < NT; ++t) b_cur[t] = b_nxt[t];
  }

  // epilogue: last k-step
#pragma unroll
  for (int t = 0; t < NT; ++t)
    acc[t] = __builtin_amdgcn_wmma_f32_16x16x4_f32(
        false, a_cur, false, b_cur[t], (short)0, acc[t], false, false);

  // C/D layout: VGPR v -> row v + 8*(lane>>4), col = ncol
  int rbase = (mt << 4) + ((lane >> 4) << 3);
#pragma unroll
  for (int t = 0; t < NT; ++t) {
    int ncol = ((ng * NT + t) << 4) + (lane & 15);
    for (int v = 0; v < 8; ++v)
      C[(size_t)(rbase + v) * Nc + ncol] = acc[t][v];
  }
}

// ------------------------------------------ attention logit per-node reduce
// h: [N, H*C] row-major; als/ald: [N*H]
__global__ void al_reduce(const float* __restrict__ h,
                          const float* __restrict__ asrc,
                          const float* __restrict__ adst,
                          float* __restrict__ als, float* __restrict__ ald,
                          int N, int H, int C) {
  int i = blockIdx.x * blockDim.x + threadIdx.x;
  if (i >= N * H) return;
  int hh = i % H;
  const float* row = h + (size_t)i * C;
  float s = 0.0f, d = 0.0f;
  for (int c = 0; c < C; ++c) {
    float v = row[c];
    s += v * asrc[hh * C + c];
    d += v * adst[hh * C + c];
  }
  als[i] = s;
  ald[i] = d;
}

__device__ __forceinline__ void edge_endpoints(const int* __restrict__ ei,
                                               long long eid, int E,
                                               int& s, int& d) {
  if (eid < (long long)E) { s = ei[eid]; d = ei[(long long)E + eid]; }
  else                    { s = d = (int)(eid - E); }   // self loop
}

// segment-max of leaky_relu(al_s[src]+al_d[dst]) into m[dst]
__global__ void edge_logit_max(const int* __restrict__ ei, int E, int N,
                               const float* __restrict__ als,
                               const float* __restrict__ ald,
                               float* __restrict__ m, int H) {
  long long idx = (long long)blockIdx.x * blockDim.x + threadIdx.x;
  long long tot = (long long)(E + N) * H;
  if (idx >= tot) return;
  int hh = (int)(idx % H);
  long long eid = idx / H;
  int s, d;
  edge_endpoints(ei, eid, E, s, d);
  float v = als[(size_t)s * H + hh] + ald[(size_t)d * H + hh];
  v = v > 0.0f ? v : NEG_SLOPE * v;
  atomicMaxFloat(&m[(size_t)d * H + hh], v);
}

// ex = exp(e - m[dst]); denom[dst] += ex
__global__ void edge_exp_sum(const int* __restrict__ ei, int E, int N,
                             const float* __restrict__ als,
                             const float* __restrict__ ald,
                             const float* __restrict__ m,
                             float* __restrict__ ex, float* __restrict__ den,
                             int H) {
  long long idx = (long long)blockIdx.x * blockDim.x + threadIdx.x;
  long long tot = (long long)(E + N) * H;
  if (idx >= tot) return;
  int hh = (int)(idx % H);
  long long eid = idx / H;
  int s, d;
  edge_endpoints(ei, eid, E, s, d);
  float v = als[(size_t)s * H + hh] + ald[(size_t)d * H + hh];
  v = v > 0.0f ? v : NEG_SLOPE * v;
  float e = expf(v - m[(size_t)d * H + hh]);
  ex[idx] = e;
  atomicAdd(&den[(size_t)d * H + hh], e);
}

// out[dst, h, c] += h[src, h, c] * ex/(den[dst,h]+eps)
// one thread per (edge, h*C+c): coalesced reads of h[src] row
__global__ void edge_scatter(const int* __restrict__ ei, int E, int N,
                             const float* __restrict__ ex,
                             const float* __restrict__ den,
                             const float* __restrict__ h,
                             float* __restrict__ out, int H, int C) {
  long long idx = (long long)blockIdx.x * blockDim.x + threadIdx.x;
  int HC = H * C;
  long long tot = (long long)(E + N) * HC;
  if (idx >= tot) return;
  int j = (int)(idx % HC);          // = hh*C + c
  long long eid = idx / HC;
  int hh = j / C;
  int s, d;
  edge_endpoints(ei, eid, E, s, d);
  float alpha = ex[eid * H + hh] / (den[(size_t)d * H + hh] + EPSV);
  atomicAdd(&out[(size_t)d * HC + j], h[(size_t)s * HC + j] * alpha);
}

// in-place: p = elu(p + bias[col])
__global__ void bias_elu(float* __restrict__ p, const float* __restrict__ b,
                         long long n, int F) {
  long long i = (long long)blockIdx.x * blockDim.x + threadIdx.x;
  if (i >= n) return;
  float v = p[i] + b[(int)(i % F)];
  p[i] = v > 0.0f ? v : (expf(v) - 1.0f);
}

__global__ void bias_add(float* __restrict__ p, const float* __restrict__ b,
                         long long n, int F) {
  long long i = (long long)blockIdx.x * blockDim.x + threadIdx.x;
  if (i >= n) return;
  p[i] += b[(int)(i % F)];
}

// ---------------------------------------------------------------- launcher

extern "C" void kernel_launch(void* const* d_in, const int* in_sizes, int n_in,
                              void* d_out, int out_size, void* d_ws, size_t ws_size,
                              hipStream_t stream) {
  const float* x      = (const float*)d_in[0];
  const int*   ei     = (const int*)  d_in[1];
  const float* W1     = (const float*)d_in[2];
  const float* a_src1 = (const float*)d_in[3];
  const float* a_dst1 = (const float*)d_in[4];
  const float* b1     = (const float*)d_in[5];
  const float* W2     = (const float*)d_in[6];
  const float* a_src2 = (const float*)d_in[7];
  const float* a_dst2 = (const float*)d_in[8];
  const float* b2     = (const float*)d_in[9];

  const int N  = in_sizes[0] / FDIM;   // 50000
  const int E  = in_sizes[1] / 2;      // 800000
  const long long Et = (long long)E + N;
  float* out = (float*)d_out;

  // workspace carve-up (floats)
  float* ws = (float*)d_ws;
  size_t off = 0;
  float* h1   = ws + off; off += (size_t)N * 256;   // layer-1 features
  float* o1   = ws + off; off += (size_t)N * 256;   // layer-1 output / elu (in-place)
  float* h2   = ws + off; off += (size_t)N * 16;    // layer-2 features
  float* als1 = ws + off; off += (size_t)N * 8;
  float* ald1 = ws + off; off += (size_t)N * 8;
  float* m1   = ws + off; off += (size_t)N * 8;
  float* den1 = ws + off; off += (size_t)N * 8;
  float* ex1  = ws + off; off += (size_t)Et * 8;
  float* als2 = ws + off; off += (size_t)N;
  float* ald2 = ws + off; off += (size_t)N;
  float* m2   = ws + off; off += (size_t)N;
  float* den2 = ws + off; off += (size_t)N;
  float* ex2  = ws + off; off += (size_t)Et;
  float* W1t  = ws + off; off += (size_t)256 * 256; // transposed weights
  float* W2t  = ws + off; off += (size_t)16 * 256;

  const int B = 256;
  #define NBLK(n) ((int)(((long long)(n) + B - 1) / B))

  // -------- init accumulators + transposed weights
  fill_f32<<<NBLK((long long)N * 16), B, 0, stream>>>(out, 0.0f, (long long)N * 16);
  fill_f32<<<NBLK((long long)N * 256), B, 0, stream>>>(o1, 0.0f, (long long)N * 256);
  fill_f32<<<NBLK((long long)N * 8), B, 0, stream>>>(den1, 0.0f, (long long)N * 8);
  fill_f32<<<NBLK((long long)N * 8), B, 0, stream>>>(m1, -INFINITY, (long long)N * 8);
  fill_f32<<<NBLK(N), B, 0, stream>>>(den2, 0.0f, (long long)N);
  fill_f32<<<NBLK(N), B, 0, stream>>>(m2, -INFINITY, (long long)N);
  transpose_f32<<<NBLK(256 * 256), B, 0, stream>>>(W1, W1t, 256, 256);
  transpose_f32<<<NBLK(256 * 16), B, 0, stream>>>(W2, W2t, 256, 16);

  // -------- layer 1: h1 = x @ W1  (fp32 WMMA, 16x64 strip per wave)
  int grp1 = (N / 16) * (256 / 64);
  gemm_f32_wmma<4><<<(grp1 + 7) / 8, 256, 0, stream>>>(x, W1t, h1, N, 256, 256);
  al_reduce<<<NBLK(N * 8), B, 0, stream>>>(h1, a_src1, a_dst1, als1, ald1, N, 8, 32);
  edge_logit_max<<<NBLK(Et * 8), B, 0, stream>>>(ei, E, N, als1, ald1, m1, 8);
  edge_exp_sum<<<NBLK(Et * 8), B, 0, stream>>>(ei, E, N, als1, ald1, m1, ex1, den1, 8);
  edge_scatter<<<NBLK(Et * 256), B, 0, stream>>>(ei, E, N, ex1, den1, h1, o1, 8, 32);
  bias_elu<<<NBLK((long long)N * 256), B, 0, stream>>>(o1, b1, (long long)N * 256, 256);

  // -------- layer 2: h2 = elu(o1) @ W2  (fp32 WMMA, 16x16 tile per wave)
  int grp2 = (N / 16) * (16 / 16);
  gemm_f32_wmma<1><<<(grp2 + 7) / 8, 256, 0, stream>>>(o1, W2t, h2, N, 256, 16);
  al_reduce<<<NBLK(N), B, 0, stream>>>(h2, a_src2, a_dst2, als2, ald2, N, 1, 16);
  edge_logit_max<<<NBLK(Et), B, 0, stream>>>(ei, E, N, als2, ald2, m2, 1);
  edge_exp_sum<<<NBLK(Et), B, 0, stream>>>(ei, E, N, als2, ald2, m2, ex2, den2, 1);
  edge_scatter<<<NBLK(Et * 16), B, 0, stream>>>(ei, E, N, ex2, den2, h2, out, 1, 16);
  bias_add<<<NBLK((long long)N * 16), B, 0, stream>>>(out, b2, (long long)N * 16, 16);

  #undef NBLK
}